// ColWarp_26843545600503
// MI455X (gfx1250) — compile-verified
//
#include <hip/hip_runtime.h>
#include <cstdint>
#include <cstddef>

#ifndef __has_builtin
#define __has_builtin(x) 0
#endif

#define IMG_H 768
#define IMG_W 768
#define TW 128            // output tile width
#define TH 16             // output tile height
#define LSTRIDE 132       // (TW+4) floats per LDS row (== TDM tile_dim0)
#define LH 20             // (TH+4) rows
#define PLANE (LSTRIDE*LH)      // 2640 floats per channel plane
#define NSLOT64 (3*LH*66)       // 3960 b64 slots for the guarded edge path
#define NPAIR (LH*66)           // 1320 float2 positions per plane

typedef unsigned int v4u __attribute__((ext_vector_type(4)));
typedef int          v8i __attribute__((ext_vector_type(8)));
typedef int          v4i __attribute__((ext_vector_type(4)));

__global__ __launch_bounds__(256)
void colwarp_dwconv_kernel(const float* __restrict__ im,
                           const float* __restrict__ fc,
                           float* __restrict__ out)
{
    __shared__ float tile[3 * PLANE];   // 31,680 B, contiguous TDM destination
    __shared__ float cst[40];

    const int tid = threadIdx.x;
    const int tx  = blockIdx.x;          // 0..5
    const int ty  = blockIdx.y;          // 0..47
    const int b   = blockIdx.z;
    const int x0  = tx * TW;
    const int y0  = ty * TH;

    // stage per-sample params (9 warp + 3 shift + 25 kernel)
    if (tid < 37) cst[tid] = fc[b * 37 + tid];

    const float* imb = im + (size_t)b * 3 * IMG_H * IMG_W;
    // TDM zero-fills the far (right/bottom) halo via tensor_dim clamping;
    // only tiles touching the left/top image border need the guarded path.
    const bool tdm_ok = (tx >= 1) && (ty >= 1);

    if (tdm_ok) {
        // ---- CDNA5 Tensor Data Mover: one 3-D tile DMA (132 x 20 x 3 fp32) ----
        if (tid < 32) {   // wave 0 issues; TDM ignores EXEC, issues once per wave
            const unsigned lds_off = (unsigned)(uintptr_t)&tile[0];
            const float* gbase = imb + (size_t)(y0 - 2) * IMG_W + (x0 - 2);
            const unsigned long long ga = (unsigned long long)(uintptr_t)gbase;
            const int dim0 = (IMG_W + 2) - x0;   // valid x elements from gbase
            const int dim1 = (IMG_H + 2) - y0;   // valid y rows from gbase

            // D# group 0: count=1 | lds_addr | global_addr[56:0] | type=2
            v4u g0 = { 1u,
                       lds_off,
                       (unsigned)(ga & 0xFFFFFFFFu),
                       (unsigned)(((ga >> 32) & 0x01FFFFFFu) | 0x80000000u) };
            // D# group 1: data_size=4B; tensor_dim0/1; tile 132x20x3;
            //             stride0=768; stride1=768*768 (0x90000)
            v8i g1 = { (int)0x00020000,
                       (int)((dim0 & 0xFFFF) << 16),
                       (int)(((unsigned)dim0 >> 16) | ((unsigned)(dim1 & 0xFFFF) << 16)),
                       (int)(((unsigned)dim1 >> 16) | (LSTRIDE << 16)),
                       (int)(LH | (3 << 16)),
                       (int)IMG_W,                    // tensor_dim0_stride[31:0]
                       0,                             // stride0[47:32]=0, stride1[15:0]=0
                       (int)((IMG_H * IMG_W) >> 16) };// stride1[47:16] = 9
            // D# group 2: tensor_dim2=3, no dim3, no iterate
            v4i g2 = { 3, 0, 0, 0 };
            v4i g3 = { 0, 0, 0, 0 };
            v8i g4 = { 0, 0, 0, 0, 0, 0, 0, 0 };   // extra arg (clang-23 form), zero

#if __has_builtin(__builtin_amdgcn_tensor_load_to_lds)
            __builtin_amdgcn_tensor_load_to_lds(g0, g1, g2, g3, g4, 0);
#else
            asm volatile("tensor_load_to_lds %0, %1, %2, %3"
                         :: "s"(g0), "s"(g1), "s"(g2), "s"(g3) : "memory");
#endif
#if __has_builtin(__builtin_amdgcn_s_wait_tensorcnt)
            __builtin_amdgcn_s_wait_tensorcnt(0);
#else
            asm volatile("s_wait_tensorcnt 0x0" ::: "memory");
#endif
        }
    } else {
        // left/top border blocks: guarded loads with zero padding
        for (int i = tid; i < NSLOT64; i += 256) {
            int ch  = i / 1320;
            int r   = i - ch * 1320;
            int row = r / 66;
            int j   = r - row * 66;
            int gy  = y0 - 2 + row;
            int gx  = x0 - 2 + 2 * j;
            float v0 = 0.0f, v1 = 0.0f;
            if ((unsigned)gy < (unsigned)IMG_H) {
                const float* gp = imb + ((size_t)(ch * IMG_H + gy)) * IMG_W;
                if ((unsigned)gx       < (unsigned)IMG_W) v0 = gp[gx];
                if ((unsigned)(gx + 1) < (unsigned)IMG_W) v1 = gp[gx + 1];
            }
            *(float2*)&tile[ch * PLANE + row * LSTRIDE + 2 * j] = make_float2(v0, v1);
        }
    }
    __syncthreads();

    // ---- in-place shift + 3x3 color warp in LDS (each value reused ~25x by conv) ----
    const float w00 = cst[0], w01 = cst[1], w02 = cst[2];
    const float w10 = cst[3], w11 = cst[4], w12 = cst[5];
    const float w20 = cst[6], w21 = cst[7], w22 = cst[8];
    const float s0 = cst[9], s1 = cst[10], s2 = cst[11];
    const float b0 = s0 * w00 + s1 * w10 + s2 * w20;
    const float b1 = s0 * w01 + s1 * w11 + s2 * w21;
    const float b2 = s0 * w02 + s1 * w12 + s2 * w22;

    auto warp_pair = [&](int i) {
        const int p = 2 * i;
        float2 r0 = *(float2*)&tile[p];
        float2 r1 = *(float2*)&tile[PLANE + p];
        float2 r2 = *(float2*)&tile[2 * PLANE + p];
        float2 o0, o1, o2;
        o0.x = b0 + r0.x * w00 + r1.x * w10 + r2.x * w20;
        o1.x = b1 + r0.x * w01 + r1.x * w11 + r2.x * w21;
        o2.x = b2 + r0.x * w02 + r1.x * w12 + r2.x * w22;
        o0.y = b0 + r0.y * w00 + r1.y * w10 + r2.y * w20;
        o1.y = b1 + r0.y * w01 + r1.y * w11 + r2.y * w21;
        o2.y = b2 + r0.y * w02 + r1.y * w12 + r2.y * w22;
        *(float2*)&tile[p]             = o0;
        *(float2*)&tile[PLANE + p]     = o1;
        *(float2*)&tile[2 * PLANE + p] = o2;
    };
#pragma unroll
    for (int it = 0; it < 5; ++it)        // 5*256 = 1280 full pairs
        warp_pair(tid + it * 256);
    if (tid < NPAIR - 5 * 256)            // remaining 40 pairs
        warp_pair(tid + 5 * 256);
    __syncthreads();

    // ---- register-tiled 5x5 depthwise conv: 8 outputs/thread/channel ----
    float kk[25];
#pragma unroll
    for (int i = 0; i < 25; ++i) kk[i] = cst[12 + i];

    const int orow = tid >> 4;       // 0..15
    const int ox   = (tid & 15) * 8; // 0..120

#pragma unroll
    for (int ch = 0; ch < 3; ++ch) {
        float acc[8];
#pragma unroll
        for (int j = 0; j < 8; ++j) acc[j] = 0.0f;

#pragma unroll
        for (int ky = 0; ky < 5; ++ky) {
            const int base = ch * PLANE + (orow + ky) * LSTRIDE + ox;
            float win[12];
            *(float4*)&win[0] = *(const float4*)&tile[base];
            *(float4*)&win[4] = *(const float4*)&tile[base + 4];
            *(float4*)&win[8] = *(const float4*)&tile[base + 8];
#pragma unroll
            for (int kx = 0; kx < 5; ++kx) {
                const float wgt = kk[ky * 5 + kx];
#pragma unroll
                for (int j = 0; j < 8; ++j)
                    acc[j] += wgt * win[j + kx];
            }
        }

        float* op = out + (((size_t)(b * 3 + ch) * IMG_H) + (y0 + orow)) * IMG_W + x0 + ox;
        *(float4*)&op[0] = make_float4(acc[0], acc[1], acc[2], acc[3]);
        *(float4*)&op[4] = make_float4(acc[4], acc[5], acc[6], acc[7]);
    }
}

extern "C" void kernel_launch(void* const* d_in, const int* in_sizes, int n_in,
                              void* d_out, int out_size, void* d_ws, size_t ws_size,
                              hipStream_t stream) {
    const float* im = (const float*)d_in[0];
    const float* fc = (const float*)d_in[1];
    float* out = (float*)d_out;
    const int bs = in_sizes[1] / 37;   // 32
    dim3 grid(IMG_W / TW, IMG_H / TH, bs);
    hipLaunchKernelGGL(colwarp_dwconv_kernel, grid, dim3(256), 0, stream, im, fc, out);
}